// TimeSeriesOps_57655640982241
// MI455X (gfx1250) — compile-verified
//
#include <hip/hip_runtime.h>

// Rolling mean, window=60, over axis 0 of (2048, 24000) fp32.
// Bandwidth-bound (~0.01 flop/byte): stream HBM once via CDNA5 async
// global->LDS loads into a ring buffer, scan from LDS, coalesced b128 I/O,
// non-temporal stores so the 196MB output stream doesn't evict L2.

#define W      60
#define BLK    120   // ceil(2.0 * W): matches reference block structure
#define PD     8     // async prefetch depth (rows in flight)
#define RING   72    // ring rows; live span = W + PD = 68 < 72
#define LANES  32    // one wave32 per workgroup, 1 float4 column per lane

typedef float v4f __attribute__((vector_size(16)));   // native vector: builtin-friendly

#if defined(__HIP_DEVICE_COMPILE__) && defined(__gfx1250__)
#  if __has_builtin(__builtin_amdgcn_global_load_async_to_lds_b128)
#    define HAVE_ASYNC 1
#  endif
#endif
#ifndef HAVE_ASYNC
#  define HAVE_ASYNC 0
#endif

#if HAVE_ASYNC
// Probe-confirmed signature:
// (v4i addrspace(1)* gaddr, v4i addrspace(3)* laddr, int offset, int cpol)
typedef int v4i __attribute__((vector_size(16)));
typedef __attribute__((address_space(1))) v4i* gv4i_p;
typedef __attribute__((address_space(3))) v4i* lv4i_p;
#  if __has_builtin(__builtin_amdgcn_s_wait_asynccnt)
#    define WAIT_ASYNC(n) __builtin_amdgcn_s_wait_asynccnt(n)
#  else
#    define WAIT_ASYNC(n) asm volatile("s_wait_asynccnt %0" ::"i"(n) : "memory")
#  endif
#else
#  define WAIT_ASYNC(n) ((void)0)
#endif

__device__ __forceinline__ void async_row(const float* __restrict__ gp,
                                          v4f* lp) {
#if HAVE_ASYNC
  __builtin_amdgcn_global_load_async_to_lds_b128(
      (gv4i_p)gp, (lv4i_p)lp, /*offset=*/0, /*cpol=*/0);
#else
  *lp = *(const v4f*)gp;      // synchronous fallback, same ring structure
#endif
}

__device__ __forceinline__ void nt_store(float* p, v4f v) {
  __builtin_nontemporal_store(v, (v4f*)p);   // store TH=NT: bypass cache residency
}

__global__ __launch_bounds__(LANES) void
TimeSeriesOps_rollmean_kernel(const float* __restrict__ data,
                              float* __restrict__ out,
                              int C4, int maxw) {
  __shared__ v4f ring[RING * LANES];

  const int tid = threadIdx.x;                 // 0..31
  int c4 = blockIdx.x * LANES + tid;           // float4 column
  const bool active = (c4 < C4);
  if (!active) c4 = C4 - 1;                    // clamp: safe loads, no stores

  const int s0 = blockIdx.y * BLK;             // first window of this block
  int nw = maxw - s0; if (nw > BLK) nw = BLK;  // windows in this block
  const int lastk = (W - 1) + (nw - 1);        // last relative row index
  const float invw = 1.0f / (float)W;

  const size_t rowstride = (size_t)C4 * 4;     // floats per row
  const float* gbase = data + (size_t)s0 * rowstride + (size_t)c4 * 4;
  float*       obase = out  + (size_t)c4 * 4;

  // ---- Phase 1: async-fill rows 0..W-1 (+PD prefetch) into the LDS ring ----
  int issued = W + PD; if (issued > lastk + 1) issued = lastk + 1;
  for (int k = 0; k < issued; ++k)
    async_row(gbase + (size_t)k * rowstride, &ring[k * LANES + tid]);
  WAIT_ASYNC(PD);   // in-order completion => rows 0..W-1 are in LDS

  v4f s = {0.f, 0.f, 0.f, 0.f};
#pragma unroll 4
  for (int k = 0; k < W; ++k)
    s += ring[k * LANES + tid];                // vector adds (4x v_add_f32)
  v4f m = s * invw;
  if (active) {
    nt_store(obase + (size_t)(s0 + W - 1) * rowstride, m);
    if (blockIdx.y == 0) {                     // replicate window 0 upward
      for (int r = 0; r < W - 1; ++r)
        nt_store(obase + (size_t)r * rowstride, m);
    }
  }

  // ---- Phase 2: incremental windows, async loads pipelined PD rows ahead ----
  int k_pref    = issued;                      // next relative row to issue
  int slot_pref = issued % RING;
  int slot_new  = W;                           // row W-1+i at i=1
  int slot_old  = 0;                           // row i-1   at i=1
  for (int i = 1; i < nw; ++i) {
    if (k_pref <= lastk) {
      async_row(gbase + (size_t)k_pref * rowstride,
                &ring[slot_pref * LANES + tid]);
      ++k_pref; if (++slot_pref == RING) slot_pref = 0;
      WAIT_ASYNC(PD);                          // row W-1+i has PD issues after it
    } else {
      WAIT_ASYNC(0);                           // tail: drain everything
    }
    v4f xi = ring[slot_new * LANES + tid];
    v4f xo = ring[slot_old * LANES + tid];
    if (++slot_new == RING) slot_new = 0;
    if (++slot_old == RING) slot_old = 0;
    s += xi - xo;
    if (active)
      nt_store(obase + (size_t)(s0 + W - 1 + i) * rowstride, s * invw);
  }
}

extern "C" void kernel_launch(void* const* d_in, const int* in_sizes, int n_in,
                              void* d_out, int out_size, void* d_ws, size_t ws_size,
                              hipStream_t stream) {
  const float* data = (const float*)d_in[0];
  float* out = (float*)d_out;

  const int T  = 2048;                 // reference setup
  const int C  = in_sizes[0] / T;      // 24000 floats per row
  const int C4 = C / 4;                // 6000 float4 columns
  const int maxw = T - W + 1;          // 1989 windows

  dim3 grid((C4 + LANES - 1) / LANES,  // 188 column blocks
            (maxw + BLK - 1) / BLK);   // 17 time blocks
  TimeSeriesOps_rollmean_kernel<<<grid, dim3(LANES), 0, stream>>>(data, out, C4, maxw);
}